// v_ProjectLayer_22746146799945
// MI455X (gfx1250) — compile-verified
//
#include <hip/hip_runtime.h>

// ProjectLayer: coarse-sampled gauge-covariant pooling, complex128 (f64) end to end.
// out[cs, s, a] = tf(x)[s,a] + sum_mu sum_b U[mu](x)[a,b] * tf(x+e_mu)[s,b]
// with tf(y)[s,c] = sum_j w(y)[s,j] * f(y)[j,c], x = 2*coarse coords (never wraps).

namespace {
constexpr int LT = 16, LX = 16, LY = 16, LZ = 32;   // fine lattice
constexpr int CT = 8,  CX = 8,  CY = 8,  CZ = 16;   // coarse lattice
constexpr int VOL = LT * LX * LY * LZ;              // 131072 fine sites
constexpr int NCS = CT * CX * CY * CZ;              // 8192 coarse sites
constexpr int NTH = NCS * 4;                        // one thread per (coarse site, spin)
constexpr int STRIDE_T = LX * LY * LZ;              // 8192
constexpr int STRIDE_X = LY * LZ;                   // 512
constexpr int STRIDE_Y = LZ;                        // 32
}

// complex fused multiply-add: acc += a * b, all lowering to v_fma_f64
__device__ __forceinline__ double2 cfma(double2 acc, const double2 a, const double2 b) {
    acc.x = fma(a.x,  b.x, acc.x);
    acc.x = fma(-a.y, b.y, acc.x);
    acc.y = fma(a.x,  b.y, acc.y);
    acc.y = fma(a.y,  b.x, acc.y);
    return acc;
}

__global__ __launch_bounds__(256)
void project_pool_f64(const double2* __restrict__ f,   // [VOL][4][3]  features (spin, color)
                      const double2* __restrict__ U,   // [4][VOL][3][3] gauge links
                      const double2* __restrict__ w,   // [VOL][4][4]  spin weights
                      double2* __restrict__ out)       // [NCS][4][3]
{
    const int tid = blockIdx.x * blockDim.x + threadIdx.x;
    if (tid >= NTH) return;

    const int s  = tid & 3;    // spin row handled by this lane
    const int cs = tid >> 2;   // coarse site

    int r = cs;
    const int cz = r % CZ; r /= CZ;
    const int cy = r % CY; r /= CY;
    const int cx = r % CX; r /= CX;
    const int ct = r;

    // fine-lattice site of this coarse site (all coords even -> +1 never wraps)
    const int s0 = (((ct << 1) * LX + (cx << 1)) * LY + (cy << 1)) * LZ + (cz << 1);
    const int sn[4] = { s0 + STRIDE_T, s0 + STRIDE_X, s0 + STRIDE_Y, s0 + 1 };

    // Kick off prefetches for all neighbor data + link rows so they fly while
    // we do the on-site spin transform (emits global_prefetch_b8 on gfx1250).
    #pragma unroll
    for (int mu = 0; mu < 4; ++mu) {
        __builtin_prefetch((const void*)(w + (size_t)sn[mu] * 16 + (size_t)s * 4), 0, 3);
        __builtin_prefetch((const void*)(f + (size_t)sn[mu] * 12), 0, 3);
        __builtin_prefetch((const void*)(U + ((size_t)mu * VOL + (size_t)s0) * 9), 0, 3);
    }

    // ---- on-site term: acc[c] = tf(s0)[s,c] ----
    double2 acc[3];
    {
        const double2* wp = w + (size_t)s0 * 16 + (size_t)s * 4; // row s of 4x4 weight
        const double2* fp = f + (size_t)s0 * 12;                 // full 4x3 feature block
        double2 wr[4], fl[12];
        #pragma unroll
        for (int j = 0; j < 4; ++j)  wr[j] = wp[j];              // global_load_b128 each
        #pragma unroll
        for (int k = 0; k < 12; ++k) fl[k] = fp[k];
        #pragma unroll
        for (int c = 0; c < 3; ++c) {
            double2 a = make_double2(0.0, 0.0);
            #pragma unroll
            for (int j = 0; j < 4; ++j) a = cfma(a, wr[j], fl[j * 3 + c]);
            acc[c] = a;
        }
    }

    // ---- hopping terms: acc[a] += sum_b U_mu(s0)[a,b] * tf(s0+e_mu)[s,b] ----
    #pragma unroll
    for (int mu = 0; mu < 4; ++mu) {
        const double2* wp = w + (size_t)sn[mu] * 16 + (size_t)s * 4;
        const double2* fp = f + (size_t)sn[mu] * 12;
        const double2* Up = U + ((size_t)mu * VOL + (size_t)s0) * 9;

        double2 wr[4], fl[12], Ur[9], tfn[3];
        #pragma unroll
        for (int j = 0; j < 4; ++j)  wr[j] = wp[j];
        #pragma unroll
        for (int k = 0; k < 12; ++k) fl[k] = fp[k];
        #pragma unroll
        for (int k = 0; k < 9; ++k)  Ur[k] = Up[k];

        #pragma unroll
        for (int c = 0; c < 3; ++c) {
            double2 a = make_double2(0.0, 0.0);
            #pragma unroll
            for (int j = 0; j < 4; ++j) a = cfma(a, wr[j], fl[j * 3 + c]);
            tfn[c] = a;
        }
        #pragma unroll
        for (int a = 0; a < 3; ++a) {
            #pragma unroll
            for (int b = 0; b < 3; ++b) acc[a] = cfma(acc[a], Ur[a * 3 + b], tfn[b]);
        }
    }

    // lanes (cs,0..3) write 4 contiguous 48B chunks -> coalesced b128 stores
    double2* op = out + (size_t)cs * 12 + (size_t)s * 3;
    op[0] = acc[0];
    op[1] = acc[1];
    op[2] = acc[2];
}

extern "C" void kernel_launch(void* const* d_in, const int* in_sizes, int n_in,
                              void* d_out, int out_size, void* d_ws, size_t ws_size,
                              hipStream_t stream) {
    // setup_inputs() order: features, U, weights  (all complex128 -> double2)
    const double2* f = (const double2*)d_in[0];
    const double2* U = (const double2*)d_in[1];
    const double2* w = (const double2*)d_in[2];
    double2* out     = (double2*)d_out;

    const int block = 256;
    const int grid  = (NTH + block - 1) / block;   // 128 blocks of 8 waves (wave32)
    hipLaunchKernelGGL(project_pool_f64, dim3(grid), dim3(block), 0, stream,
                       f, U, w, out);
}